// SpikingVestibular_16673063043180
// MI455X (gfx1250) — compile-verified
//
#include <hip/hip_runtime.h>
#include <stdint.h>

// Problem constants (match reference)
#define B_     1000000
#define N_     6
#define BN_    (B_ * N_)        // 6,000,000 elements
#define STEPS_ 10
#define TPB    256              // 8 wave32 per block
#define EPT    4                // elements per thread (float4)
#define EPB    (TPB * EPT)      // 1024 elements per block
#define SLAB_BYTES (EPB * 4)    // 4096 B LDS per step slab

typedef float v4f __attribute__((ext_vector_type(4)));

// Drain slab S from LDS and advance the neuron state one step.
// Partial wait: async loads complete in order, so waiting ASYNCcnt <= (9-S)
// guarantees slabs 0..S have landed in LDS while later loads stay in flight.
template <int S>
__device__ __forceinline__ void drain_and_step(const float* __restrict__ buf, int tid,
                                               const float (&Ib)[EPT], float (&v)[EPT],
                                               float (&u)[EPT], float (&r)[EPT]) {
    if constexpr (S < STEPS_) {
#if defined(__HIP_DEVICE_COMPILE__) && defined(__gfx1250__)
        asm volatile("s_wait_asynccnt %0" :: "n"(STEPS_ - 1 - S) : "memory");
#endif
        const v4f nz = *(const v4f*)&buf[S * EPB + tid * EPT];  // ds_load_b128
        const float eps[EPT] = {nz.x, nz.y, nz.z, nz.w};
        #pragma unroll
        for (int j = 0; j < EPT; ++j) {
            const float Iin = Ib[j] + eps[j] * 0.3f;
            float x = v[j];
            x = x + (0.04f * x * x + 5.0f * x + 140.0f - u[j] + Iin);
            const float un = u[j] + 0.02f * (0.2f * x - u[j]);
            const bool sp = (x >= 30.0f);
            v[j] = sp ? -65.0f : x;
            u[j] = un + (sp ? 8.0f : 0.0f);
            r[j] = r[j] + 0.1f * ((sp ? 1.0f : 0.0f) - r[j]);
        }
        drain_and_step<S + 1>(buf, tid, Ib, v, u, r);
    }
}

__global__ __launch_bounds__(TPB) void izhikevich_rate_kernel(
    const float* __restrict__ speed,
    const float* __restrict__ turn,
    const float* __restrict__ noise,   // [STEPS, B, N] flat
    const float* __restrict__ v0,
    const float* __restrict__ u0,
    const float* __restrict__ r0,
    float* __restrict__ out)           // [B, N] flat
{
    __shared__ float buf[STEPS_ * EPB];   // 40 KB staging for noise slabs

    const int tid = (int)threadIdx.x;
    const int idx = ((int)blockIdx.x * TPB + tid) * EPT;  // first flat element
    if (idx >= BN_) return;

    // ---- Kick off 10 async B128 loads: noise[s, idx..idx+3] -> LDS slab s ----
    // NT hint: 240 MB single-use stream must not wash the 192 MB L2.
    // Low 32 bits of a shared-aperture generic address == workgroup LDS offset.
    const uint32_t lds_base = (uint32_t)(uintptr_t)(void*)buf + (uint32_t)tid * 16u;
#if defined(__HIP_DEVICE_COMPILE__) && defined(__gfx1250__)
    #pragma unroll
    for (int s = 0; s < STEPS_; ++s) {
        uint64_t ga = (uint64_t)(uintptr_t)(noise + (size_t)s * BN_ + idx);
        uint32_t la = lds_base + (uint32_t)(s * SLAB_BYTES);
        asm volatile("global_load_async_to_lds_b128 %0, %1, off th:TH_LOAD_NT"
                     :: "v"(la), "v"(ga)
                     : "memory");
    }
#endif

    // ---- While loads fly: state init (NT loads, single-use streams) ----
    v4f vv = __builtin_nontemporal_load((const v4f*)(v0 + idx));
    v4f uu = __builtin_nontemporal_load((const v4f*)(u0 + idx));
    v4f rr = __builtin_nontemporal_load((const v4f*)(r0 + idx));

    float v[EPT] = {vv.x, vv.y, vv.z, vv.w};
    float u[EPT] = {uu.x, uu.y, uu.z, uu.w};
    float r[EPT] = {rr.x, rr.y, rr.z, rr.w};

    // ---- Per-element input current (I - 1.0 tonic), from per-batch scalars ----
    float Ib[EPT];
    #pragma unroll
    for (int j = 0; j < EPT; ++j) {
        const unsigned e = (unsigned)(idx + j);
        const unsigned b = e / 6u;
        const unsigned n = e - b * 6u;
        const float tr = turn[b];     // cached: reused by 6 channels
        const float sp = speed[b];
        const float tilt = fminf(1.0f, fabsf(tr) * sp * 0.5f);
        float I;
        switch (n) {
            case 0:  I = fmaxf(0.0f,  tr) * 10.0f;        break;
            case 1:  I = fmaxf(0.0f, -tr) * 10.0f;        break;
            case 2:  I = sp * 5.0f;                       break;
            case 3:  I = fmaxf(0.0f, 0.5f - sp) * 5.0f;   break;
            default: I = tilt * 8.0f;                     break;  // n == 4, 5
        }
        Ib[j] = I - 1.0f;   // fold in I_TONIC
    }

    // ---- Pipelined drain: wait asynccnt<=(9-s), ds_load slab s, compute ----
    drain_and_step<0>(buf, tid, Ib, v, u, r);

    // ---- Store firing rate (NT: written once, never re-read) ----
    v4f ro;
    ro.x = r[0]; ro.y = r[1]; ro.z = r[2]; ro.w = r[3];
    __builtin_nontemporal_store(ro, (v4f*)(out + idx));
}

extern "C" void kernel_launch(void* const* d_in, const int* in_sizes, int n_in,
                              void* d_out, int out_size, void* d_ws, size_t ws_size,
                              hipStream_t stream) {
    // setup_inputs order: heading(0, unused), speed(1), turn_rate(2), noise(3),
    //                     v0(4), u0(5), rate0(6)
    const float* speed = (const float*)d_in[1];
    const float* turn  = (const float*)d_in[2];
    const float* noise = (const float*)d_in[3];
    const float* v0    = (const float*)d_in[4];
    const float* u0    = (const float*)d_in[5];
    const float* r0    = (const float*)d_in[6];
    float* out = (float*)d_out;

    const int blocks = (BN_ + EPB - 1) / EPB;   // 5860
    izhikevich_rate_kernel<<<blocks, TPB, 0, stream>>>(speed, turn, noise, v0, u0, r0, out);
}